// FusedDecoder_58480274703244
// MI455X (gfx1250) — compile-verified
//
#include <hip/hip_runtime.h>
#include <hip/hip_bf16.h>
#include <math.h>

// ---------------------------------------------------------------------------
// SO(3) spherical CNN decoder for MI455X (gfx1250, wave32).
// Heavy einsums (Yl, Zl) run as complex GEMMs on V_WMMA_F32_16X16X4_F32
// (full f32 precision on the matrix pipe). All constants (quad weights,
// Wigner-d via matrix exponential, grid phases) are built on device each call.
// Workspace budget ~400 MB (hot per-l GEMM operands fit in the 192MB L2).
// ---------------------------------------------------------------------------

#define PI_F 3.14159265358979323846f

typedef __attribute__((ext_vector_type(2))) float v2f;
typedef __attribute__((ext_vector_type(8))) float v8f;

// Only the real device-codegen pass sees the amdgcn builtin; the host pass
// (which still parses __global__ bodies) gets a trivial stand-in.
__device__ __forceinline__ v8f wmma_f32_4(v2f a, v2f b, v8f c)
{
#if defined(__HIP_DEVICE_COMPILE__) && __HIP_DEVICE_COMPILE__
    return __builtin_amdgcn_wmma_f32_16x16x4_f32(false, a, false, b, (short)0, c, false, false);
#else
    (void)a; (void)b;
    return c;
#endif
}

// ---------------------------------------------------------------------------
// Constants: DH betas, quadrature weights, grid alpha/gamma, grid betas
// ---------------------------------------------------------------------------
__global__ void k_init_consts(float* betaS0, float* betaS1, float* betaS2,
                              float* betaG64, float* betaG128,
                              float* w5, float* w20,
                              float* galpha, float* ggamma)
{
    int t = threadIdx.x;
    if (t < 10) betaS0[t] = PI_F * (2 * t + 1) / (4.0f * 5.0f);
    if (t < 40) betaS1[t] = PI_F * (2 * t + 1) / (4.0f * 20.0f);
    if (t < 80) betaS2[t] = PI_F * (2 * t + 1) / (4.0f * 40.0f);
    if (t == 0) { betaG64[0] = PI_F / 64.0f; betaG128[0] = PI_F / 128.0f; }
    if (t < 10) {
        const int b = 5; float s = 0.f;
        for (int k = 0; k < b; ++k)
            s += sinf((2 * t + 1) * (2 * k + 1) * PI_F / (4 * b)) / (float)(2 * k + 1);
        w5[t] = 2.0f / b * sinf(PI_F * (2 * t + 1) / (4 * b)) * s;
    }
    if (t < 40) {
        const int b = 20; float s = 0.f;
        for (int k = 0; k < b; ++k)
            s += sinf((2 * t + 1) * (2 * k + 1) * PI_F / (4 * b)) / (float)(2 * k + 1);
        w20[t] = 2.0f / b * sinf(PI_F * (2 * t + 1) / (4 * b)) * s;
    }
    if (t < 36) {
        int ai = t / 6, gj = t % 6;
        float alpha = 2.0f * PI_F * (float)ai / 6.0f;
        float pg = -2.0f * PI_F + (float)gj * (4.0f * PI_F / 6.0f);
        galpha[t] = alpha;
        ggamma[t] = pg - alpha;     // gamma = pre_gamma - alpha
    }
}

// ---------------------------------------------------------------------------
// Wigner-d:  d_l(beta) = exp( beta * (Jp^T - Jp)/2 ), real antisymmetric
// tridiagonal generator.  Scaling-and-squaring (2^-8, 12 Taylor terms, 8
// squarings) done cooperatively in LDS.  One block per beta.
// out layout: [j][m][k], n = 2l+1.
// ---------------------------------------------------------------------------
#define NW 40
__global__ __launch_bounds__(256)
void k_wigner(float* __restrict__ out, const float* __restrict__ betas, int l)
{
    __shared__ float sm[4 * NW * NW];
    float* A = sm;
    float* R = sm + NW * NW;
    float* T = sm + 2 * NW * NW;
    float* U = sm + 3 * NW * NW;
    const int n = 2 * l + 1;
    const int nn = n * n;
    const int tid = threadIdx.x, nt = blockDim.x;
    const float beta = betas[blockIdx.x];
    const float bp = beta * (1.0f / 256.0f);

    for (int e = tid; e < nn; e += nt) {
        float idv = ((e % (n + 1)) == 0) ? 1.f : 0.f;
        A[e] = 0.f; R[e] = idv; T[e] = idv;
    }
    __syncthreads();
    for (int t = tid; t < n - 1; t += nt) {
        float m = (float)(t - l);
        float c = sqrtf((float)(l * (l + 1)) - m * (m + 1.f));
        A[t * n + t + 1]   =  0.5f * bp * c;
        A[(t + 1) * n + t] = -0.5f * bp * c;
    }
    __syncthreads();
    for (int term = 1; term <= 12; ++term) {
        for (int e = tid; e < nn; e += nt) {
            int r = e / n, c = e % n; float acc = 0.f;
            for (int k = 0; k < n; ++k) acc += T[r * n + k] * A[k * n + c];
            U[e] = acc;
        }
        __syncthreads();
        float inv = 1.f / (float)term;
        for (int e = tid; e < nn; e += nt) { float v = U[e] * inv; T[e] = v; R[e] += v; }
        __syncthreads();
    }
    for (int s = 0; s < 8; ++s) {
        for (int e = tid; e < nn; e += nt) {
            int r = e / n, c = e % n; float acc = 0.f;
            for (int k = 0; k < n; ++k) acc += R[r * n + k] * R[k * n + c];
            U[e] = acc;
        }
        __syncthreads();
        for (int e = tid; e < nn; e += nt) R[e] = U[e];
        __syncthreads();
    }
    for (int e = tid; e < nn; e += nt) out[(size_t)blockIdx.x * nn + e] = R[e];
}

// ---------------------------------------------------------------------------
// Elementwise / transform helpers
// ---------------------------------------------------------------------------
__global__ void k_zero(float* p, int n)
{
    int i = blockIdx.x * blockDim.x + threadIdx.x;
    if (i < n) p[i] = 0.f;
}

__global__ void k_r2c(const float* __restrict__ in, float2* __restrict__ out, int n)
{
    int i = blockIdx.x * blockDim.x + threadIdx.x;
    if (i < n) { float2 r; r.x = in[i]; r.y = 0.f; out[i] = r; }
}

// Direct DFT along the middle axis of a [outer][N][inner] complex array.
// sign=+1, scale=1/N per axis -> ifft;  sign=-1, scale=1 -> fft.
__global__ void k_dft(const float2* __restrict__ in, float2* __restrict__ out,
                      int outer, int N, int inner, float sign, float scale)
{
    int idx = blockIdx.x * blockDim.x + threadIdx.x;
    int total = outer * N * inner;
    if (idx >= total) return;
    int i = idx % inner;
    int t = idx / inner;
    int p = t % N;
    int o = t / N;
    const float w0 = sign * 2.0f * PI_F / (float)N;
    float re = 0.f, im = 0.f;
    for (int a = 0; a < N; ++a) {
        float2 v = in[((size_t)o * N + a) * inner + i];
        float ang = w0 * (float)(p * a);
        float c = cosf(ang), s = sinf(ang);
        re += v.x * c - v.y * s;
        im += v.x * s + v.y * c;
    }
    float2 r; r.x = re * scale; r.y = im * scale;
    out[idx] = r;
}

// ---------------------------------------------------------------------------
// Xl[m][(i,k)] = sum_j w[j] * d_in[j,m,k] * xi[i, j, (m-l)%nbi, (k-l)%nbi]
// Written as GEMM-A planes (M = n, K = fin*n).
// ---------------------------------------------------------------------------
__global__ void k_Xl(const float2* __restrict__ xi, const float* __restrict__ din,
                     const float* __restrict__ w, int nbi, int fin, int l,
                     float* __restrict__ Are, float* __restrict__ Aim)
{
    const int n = 2 * l + 1;
    int idx = blockIdx.x * blockDim.x + threadIdx.x;
    int total = n * n * fin;
    if (idx >= total) return;
    int i = idx % fin;
    int t = idx / fin;
    int k = t % n;
    int m = t / n;
    int am = (m - l) % nbi; if (am < 0) am += nbi;
    int ak = (k - l) % nbi; if (ak < 0) ak += nbi;
    float re = 0.f, im = 0.f;
    for (int j = 0; j < nbi; ++j) {
        float d = w[j] * din[((size_t)j * n + m) * n + k];
        float2 v = xi[(((size_t)i * nbi + j) * nbi + am) * nbi + ak];
        re += d * v.x;
        im += d * v.y;
    }
    size_t off = (size_t)m * fin * n + (size_t)i * n + k;
    Are[off] = re; Aim[off] = im;
}

// conj(grid_D)[g][(k,n)] = e^{+i(alpha_g*(k-l) + gamma_g*(nc-l))} * d(beta_g)[k,nc]
// Written as GEMM-B planes (K = 36, N = n*n).
__global__ void k_gridD(const float* __restrict__ dbeta, const float* __restrict__ ga,
                        const float* __restrict__ gg, int l,
                        float* __restrict__ Bre, float* __restrict__ Bim)
{
    const int n = 2 * l + 1;
    int idx = blockIdx.x * blockDim.x + threadIdx.x;
    int total = 36 * n * n;
    if (idx >= total) return;
    int nc = idx % n;
    int t = idx / n;
    int k = t % n;
    int g = t / n;
    float ang = ga[g] * (float)(k - l) + gg[g] * (float)(nc - l);
    float d = dbeta[k * n + nc];
    Bre[idx] = d * cosf(ang);
    Bim[idx] = d * sinf(ang);
}

// ---------------------------------------------------------------------------
// Complex GEMM  C = alpha * A * B  on V_WMMA_F32_16X16X4_F32 (wave32).
// SoA planes; Ai == nullptr means A is real (saves half the WMMAs).
// One wave computes one 16x16 C tile. f32 WMMA has no A/B negate, so the
// real part is combined as (ArBr - AiBi) after the K loop.
// A frag (16x4): lane=16*kh+m; vgpr v holds K = k0 + 2*kh + v (contiguous,
//   8B-aligned when K%4==0 -> single b64 load in the fast path).
// B frag (4x16): lane=16*kh+n; vgpr v holds row K = k0 + 2*kh + v.
// C frag: vgpr r -> row r (lanes 0-15) / r+8 (lanes 16-31).
// Fast path (wave-uniform): interior tile + K%4==0 -> pointer-stepped,
// unguarded loads/stores. Guarded generic path only for edge tiles.
// ---------------------------------------------------------------------------
__global__ __launch_bounds__(128)
void k_cgemm_wmma(const float* __restrict__ Ar, const float* __restrict__ Ai,
                  const float* __restrict__ Br, const float* __restrict__ Bi,
                  float* __restrict__ Cr, float* __restrict__ Ci,
                  int M, int K, int N, float alpha)
{
    const int lane = threadIdx.x & 31;
    const int wid  = threadIdx.x >> 5;
    const int ntn  = (N + 15) >> 4;
    const int ntm  = (M + 15) >> 4;
    const int tile = blockIdx.x * 4 + wid;
    if (tile >= ntm * ntn) return;               // uniform per wave
    const int tm = tile / ntn, tn = tile % ntn;
    const int row0 = tm << 4, col0 = tn << 4;
    const int l16 = lane & 15;
    const int kh  = lane >> 4;
    const bool hasAi = (Ai != nullptr);

    v8f rr = {0.f,0.f,0.f,0.f,0.f,0.f,0.f,0.f};
    v8f ri = {0.f,0.f,0.f,0.f,0.f,0.f,0.f,0.f};
    v8f ii = {0.f,0.f,0.f,0.f,0.f,0.f,0.f,0.f};
    v8f ir = {0.f,0.f,0.f,0.f,0.f,0.f,0.f,0.f};

    const int arow = row0 + l16;
    const int bcol = col0 + l16;
    const bool interior = (row0 + 16 <= M) && (col0 + 16 <= N) && ((K & 3) == 0);

    if (interior) {
        // Pointer-stepped hot loop: no per-iteration index multiplies, no
        // bounds checks, EXEC stays all-ones throughout.
        const float* ap  = Ar + (size_t)arow * K + 2 * kh;
        const float* aip = hasAi ? (Ai + (size_t)arow * K + 2 * kh) : ap;
        const float* brp = Br + (size_t)(2 * kh) * N + bcol;
        const float* bip = Bi + (size_t)(2 * kh) * N + bcol;
        const size_t bstep = (size_t)4 * N;
        for (int k0 = 0; k0 < K; k0 += 4) {
            __builtin_prefetch(brp + 2 * bstep, 0, 1);
            v2f fa_r = *(const v2f*)ap;                  // aligned b64 load
            v2f fb_r; fb_r[0] = brp[0]; fb_r[1] = brp[N];
            v2f fb_i; fb_i[0] = bip[0]; fb_i[1] = bip[N];
            rr = wmma_f32_4(fa_r, fb_r, rr);
            ri = wmma_f32_4(fa_r, fb_i, ri);
            if (hasAi) {
                v2f fa_i = *(const v2f*)aip;
                ii = wmma_f32_4(fa_i, fb_i, ii);
                ir = wmma_f32_4(fa_i, fb_r, ir);
            }
            ap += 4; aip += 4; brp += bstep; bip += bstep;
        }
#pragma unroll
        for (int r = 0; r < 8; ++r) {
            const size_t o = (size_t)(row0 + r + 8 * kh) * N + bcol;
            Cr[o] = alpha * (rr[r] - ii[r]);
            Ci[o] = alpha * (ri[r] + ir[r]);
        }
    } else {
        for (int k0 = 0; k0 < K; k0 += 4) {
            v2f fa_r, fa_i, fb_r, fb_i;
#pragma unroll
            for (int v = 0; v < 2; ++v) {
                const int kk = k0 + 2 * kh + v;
                const bool aok = (arow < M) && (kk < K);
                const bool bok = (kk < K) && (bcol < N);
                fa_r[v] = aok ? Ar[(size_t)arow * K + kk] : 0.f;
                fa_i[v] = (aok && hasAi) ? Ai[(size_t)arow * K + kk] : 0.f;
                fb_r[v] = bok ? Br[(size_t)kk * N + bcol] : 0.f;
                fb_i[v] = bok ? Bi[(size_t)kk * N + bcol] : 0.f;
            }
            rr = wmma_f32_4(fa_r, fb_r, rr);
            ri = wmma_f32_4(fa_r, fb_i, ri);
            if (hasAi) {
                ii = wmma_f32_4(fa_i, fb_i, ii);
                ir = wmma_f32_4(fa_i, fb_r, ir);
            }
        }
#pragma unroll
        for (int r = 0; r < 8; ++r) {
            const int row = row0 + r + 8 * kh;
            if (row < M && bcol < N) {
                Cr[(size_t)row * N + bcol] = alpha * (rr[r] - ii[r]);
                Ci[(size_t)row * N + bcol] = alpha * (ri[r] + ir[r]);
            }
        }
    }
}

// Repack Ytmp[(i,o)][(k,nc)] -> BZ[(i,k)][(o,nc)]  (planes)
__global__ void k_repack(const float* __restrict__ Yr, const float* __restrict__ Yi,
                         float* __restrict__ BZr, float* __restrict__ BZi,
                         int fin, int fout, int n)
{
    int idx = blockIdx.x * blockDim.x + threadIdx.x;
    int total = fin * fout * n * n;
    if (idx >= total) return;
    int nc = idx % n;
    int t = idx / n;
    int k = t % n;
    t /= n;
    int o = t % fout;
    int i = t / fout;
    size_t dst = (size_t)(i * n + k) * (fout * n) + (size_t)o * n + nc;
    BZr[dst] = Yr[idx];
    BZi[dst] = Yi[idx];
}

// fhat[o, j, (m-l)%nbo, (nc-l)%nbo] += (2l+1) * d_out[j,m,nc] * Zl[m][(o,nc)]
__global__ void k_contrib(const float* __restrict__ Zr, const float* __restrict__ Zi,
                          const float* __restrict__ dout, float2* __restrict__ fhat,
                          int l, int nbo, int fout)
{
    const int n = 2 * l + 1;
    int idx = blockIdx.x * blockDim.x + threadIdx.x;
    int total = fout * nbo * n * n;
    if (idx >= total) return;
    int nc = idx % n;
    int t = idx / n;
    int m = t % n;
    t /= n;
    int j = t % nbo;
    int o = t / nbo;
    float d = (float)(2 * l + 1) * dout[((size_t)j * n + m) * n + nc];
    int om = (m - l) % nbo;  if (om < 0) om += nbo;
    int on = (nc - l) % nbo; if (on < 0) on += nbo;
    size_t zoff = (size_t)m * fout * n + (size_t)o * n + nc;
    size_t fo = (((size_t)o * nbo + j) * nbo + om) * nbo + on;
    fhat[fo].x += d * Zr[zoff];
    fhat[fo].y += d * Zi[zoff];
}

__global__ void k_finalize(const float2* __restrict__ fh, const float* __restrict__ bias,
                           float* __restrict__ y, int C, int S)
{
    int idx = blockIdx.x * blockDim.x + threadIdx.x;
    if (idx >= C * S) return;
    y[idx] = fh[idx].x + bias[idx / S];
}

__global__ __launch_bounds__(256)
void k_bnstats(const float* __restrict__ y, int S, float* __restrict__ mu, float* __restrict__ var)
{
    __shared__ float s1[256], s2[256];
    const int c = blockIdx.x;
    float a = 0.f, b = 0.f;
    for (int i = threadIdx.x; i < S; i += 256) {
        float v = y[(size_t)c * S + i];
        a += v; b += v * v;
    }
    s1[threadIdx.x] = a; s2[threadIdx.x] = b;
    __syncthreads();
    for (int st = 128; st > 0; st >>= 1) {
        if (threadIdx.x < st) { s1[threadIdx.x] += s1[threadIdx.x + st]; s2[threadIdx.x] += s2[threadIdx.x + st]; }
        __syncthreads();
    }
    if (threadIdx.x == 0) {
        float m = s1[0] / (float)S;
        mu[c] = m;
        var[c] = s2[0] / (float)S - m * m;
    }
}

__global__ void k_bn_prelu(float* __restrict__ y, const float* __restrict__ g,
                           const float* __restrict__ be, const float* __restrict__ mu,
                           const float* __restrict__ var, const float* __restrict__ a,
                           int C, int S)
{
    int idx = blockIdx.x * blockDim.x + threadIdx.x;
    if (idx >= C * S) return;
    int c = idx / S;
    float v = y[idx];
    v = g[c] * (v - mu[c]) * rsqrtf(var[c] + 1e-5f) + be[c];
    if (a) { float aa = a[0]; v = (v >= 0.f) ? v : aa * v; }
    y[idx] = v;
}

// nearest-neighbor unpool x4 in each of the three spatial dims (10 -> 40)
__global__ void k_unpool(const float* __restrict__ in, float* __restrict__ out, int total)
{
    int idx = blockIdx.x * blockDim.x + threadIdx.x;
    if (idx >= total) return;
    int Q = idx % 40;
    int t = idx / 40;
    int P = t % 40;
    t /= 40;
    int J = t % 40;
    int f = t / 40;
    out[idx] = in[(((size_t)f * 10 + J / 4) * 10 + P / 4) * 10 + Q / 4];
}

// out[(o,j,p)] = (2*pi/80) * sum_q h[(o,j,p,q)]
__global__ void k_sumlast(const float* __restrict__ in, float* __restrict__ out, int total, int n)
{
    int idx = blockIdx.x * blockDim.x + threadIdx.x;
    if (idx >= total) return;
    float s = 0.f;
    const float* p = in + (size_t)idx * n;
    for (int q = 0; q < n; ++q) s += p[q];
    out[idx] = s * (2.0f * PI_F / (float)n);
}

// ---------------------------------------------------------------------------
// Host orchestration
// ---------------------------------------------------------------------------
extern "C" void kernel_launch(void* const* d_in, const int* in_sizes, int n_in,
                              void* d_out, int out_size, void* d_ws, size_t ws_size,
                              hipStream_t stream)
{
    const float* x   = (const float*)d_in[0];
    const float* k1  = (const float*)d_in[1];
    const float* b1  = (const float*)d_in[2];
    const float* g1  = (const float*)d_in[3];
    const float* be1 = (const float*)d_in[4];
    const float* k2  = (const float*)d_in[5];
    const float* b2  = (const float*)d_in[6];
    const float* g2  = (const float*)d_in[7];
    const float* be2 = (const float*)d_in[8];
    const float* a1  = (const float*)d_in[9];
    const float* k3  = (const float*)d_in[10];
    const float* b3  = (const float*)d_in[11];
    const float* g3  = (const float*)d_in[12];
    const float* be3 = (const float*)d_in[13];
    const float* a2  = (const float*)d_in[14];
    float* outp = (float*)d_out;

    // --- workspace allocator (float units, 256B aligned) ---
    char* wsb = (char*)d_ws;
    size_t off = 0;
    auto alloc = [&](size_t nfloats) -> float* {
        float* p = (float*)(wsb + off);
        off += ((nfloats * sizeof(float) + 255) / 256) * 256;
        return p;
    };
    auto wigCum = [](int l) -> size_t {
        size_t s = 0;
        for (int t = 0; t < l; ++t) { size_t n = 2 * t + 1; s += n * n; }
        return s;
    };

    float* betaS0  = alloc(16);
    float* betaS1  = alloc(48);
    float* betaS2  = alloc(96);
    float* betaG64 = alloc(8);
    float* betaG128= alloc(8);
    float* w5      = alloc(16);
    float* w20     = alloc(48);
    float* galpha  = alloc(64);
    float* ggamma  = alloc(64);
    float* wigS0   = alloc(10 * wigCum(5));    // b=5 betas,  l<5
    float* wigS1   = alloc(40 * wigCum(20));   // b=20 betas, l<20
    float* wigS2   = alloc(80 * wigCum(20));   // b=40 betas, l<20
    float* wigG64  = alloc(wigCum(5));         // beta=pi/64
    float* wigG128 = alloc(wigCum(20));        // beta=pi/128
    float* XI      = alloc(8192000);           // xi, complex interleaved (max 64*40^3)
    float* FH      = alloc(32768000);          // fhat, complex interleaved (max 32*80^3)
    float* Ubase   = alloc(32768000);          // union: DFT tmp | Y/BZ planes
    float* Ytr = Ubase;
    float* Yti = Ubase + 6250000;
    float* BZr = Ubase + 12500000;
    float* BZi = Ubase + 18750000;
    float* Are = alloc(100000);
    float* Aim = alloc(100000);
    float* Dre = alloc(60000);
    float* Dim = alloc(60000);
    float* Zr  = alloc(100000);
    float* Zi  = alloc(100000);
    float* d1  = alloc(64000);                 // layer-1 out (64,10,10,10)
    float* ub  = alloc(4096000);               // unpooled (64,40,40,40)
    float* h2  = alloc(4096000);               // layer-2 out
    float* y3  = alloc(16384000);              // layer-3 out (32,80,80,80)
    float* muB = alloc(64);
    float* varB= alloc(64);
    (void)ws_size; (void)in_sizes; (void)n_in; (void)out_size;

    // --- constants ---
    k_init_consts<<<1, 128, 0, stream>>>(betaS0, betaS1, betaS2, betaG64, betaG128,
                                         w5, w20, galpha, ggamma);
    for (int l = 0; l < 5;  ++l) k_wigner<<<10, 256, 0, stream>>>(wigS0 + 10 * wigCum(l), betaS0, l);
    for (int l = 0; l < 20; ++l) k_wigner<<<40, 256, 0, stream>>>(wigS1 + 40 * wigCum(l), betaS1, l);
    for (int l = 0; l < 20; ++l) k_wigner<<<80, 256, 0, stream>>>(wigS2 + 80 * wigCum(l), betaS2, l);
    for (int l = 0; l < 5;  ++l) k_wigner<<<1, 256, 0, stream>>>(wigG64 + wigCum(l), betaG64, l);
    for (int l = 0; l < 20; ++l) k_wigner<<<1, 256, 0, stream>>>(wigG128 + wigCum(l), betaG128, l);

    auto run_layer = [&](const float* xin, int fin, int fout, int b_in, int b_out, int b_inv,
                         const float* kern, const float* bias, const float* bng, const float* bnb,
                         const float* pa, const float* wq,
                         const float* wigIn, int nbIn, const float* wigOut, int nbOut,
                         const float* wigGrid, float* yout)
    {
        const int nbi = 2 * b_in, nbo = 2 * b_inv;
        const int lmax = (b_out < b_inv) ? b_out : b_inv;
        const float scaling = 1.0f / sqrtf(36.0f * (float)fin *
                              ((float)b_out * b_out * b_out) / ((float)b_in * b_in * b_in));
        float2* XI2  = (float2*)XI;
        float2* FH2  = (float2*)FH;
        float2* TMP2 = (float2*)Ubase;

        // xi = ifft2(x) over (alpha,gamma)
        {
            int tot = fin * nbi * nbi * nbi;
            k_r2c<<<(tot + 255) / 256, 256, 0, stream>>>(xin, XI2, tot);
            k_dft<<<(tot + 255) / 256, 256, 0, stream>>>(XI2, TMP2, fin * nbi * nbi, nbi, 1,   +1.f, 1.f / nbi);
            k_dft<<<(tot + 255) / 256, 256, 0, stream>>>(TMP2, XI2, fin * nbi,       nbi, nbi, +1.f, 1.f / nbi);
        }
        {
            int tot = fout * nbo * nbo * nbo * 2;
            k_zero<<<(tot + 255) / 256, 256, 0, stream>>>(FH, tot);
        }

        for (int l = 0; l < lmax; ++l) {
            const int n = 2 * l + 1;
            const float* dIn  = wigIn  + (size_t)nbIn  * wigCum(l);
            const float* dOut = wigOut + (size_t)nbOut * wigCum(l);
            const float* dG   = wigGrid + wigCum(l);

            { int tot = n * n * fin;
              k_Xl<<<(tot + 255) / 256, 256, 0, stream>>>(XI2, dIn, wq, nbi, fin, l, Are, Aim); }
            { int tot = 36 * n * n;
              k_gridD<<<(tot + 255) / 256, 256, 0, stream>>>(dG, galpha, ggamma, l, Dre, Dim); }
            // Yl: (fin*fout x 36) * (36 x n^2), real A (kernel) x complex B
            { int M = fin * fout, K = 36, N = n * n;
              int tiles = ((M + 15) / 16) * ((N + 15) / 16);
              k_cgemm_wmma<<<(tiles + 3) / 4, 128, 0, stream>>>(kern, nullptr, Dre, Dim,
                                                                Ytr, Yti, M, K, N, scaling); }
            { int tot = fin * fout * n * n;
              k_repack<<<(tot + 255) / 256, 256, 0, stream>>>(Ytr, Yti, BZr, BZi, fin, fout, n); }
            // Zl: (n x fin*n) * (fin*n x fout*n), complex x complex
            { int M = n, K = fin * n, N = fout * n;
              int tiles = ((M + 15) / 16) * ((N + 15) / 16);
              k_cgemm_wmma<<<(tiles + 3) / 4, 128, 0, stream>>>(Are, Aim, BZr, BZi,
                                                                Zr, Zi, M, K, N, 1.0f); }
            { int tot = fout * nbo * n * n;
              k_contrib<<<(tot + 255) / 256, 256, 0, stream>>>(Zr, Zi, dOut, FH2, l, nbo, fout); }
        }

        // y = Re(fft2(fhat)) + bias ; batchnorm ; optional prelu
        {
            int tot = fout * nbo * nbo * nbo;
            k_dft<<<(tot + 255) / 256, 256, 0, stream>>>(FH2, TMP2, fout * nbo * nbo, nbo, 1,   -1.f, 1.f);
            k_dft<<<(tot + 255) / 256, 256, 0, stream>>>(TMP2, FH2, fout * nbo,       nbo, nbo, -1.f, 1.f);
            k_finalize<<<(tot + 255) / 256, 256, 0, stream>>>(FH2, bias, yout, fout, nbo * nbo * nbo);
            k_bnstats<<<fout, 256, 0, stream>>>(yout, nbo * nbo * nbo, muB, varB);
            k_bn_prelu<<<(tot + 255) / 256, 256, 0, stream>>>(yout, bng, bnb, muB, varB, pa,
                                                              fout, nbo * nbo * nbo);
        }
    };

    // layer 1: (1,128,10,10,10) -> (1,64,10,10,10), bn, no prelu
    run_layer(x, 128, 64, 5, 5, 5, k1, b1, g1, be1, nullptr, w5,
              wigS0, 10, wigS0, 10, wigG64, d1);
    // unpool x4 -> (1,64,40,40,40)
    { int tot = 64 * 40 * 40 * 40;
      k_unpool<<<(tot + 255) / 256, 256, 0, stream>>>(d1, ub, tot); }
    // layer 2: (1,64,40,40,40) -> same, bn + prelu(a1)
    run_layer(ub, 64, 64, 20, 20, 20, k2, b2, g2, be2, a1, w20,
              wigS1, 40, wigS1, 40, wigG128, h2);
    // layer 3: (1,64,40,40,40) -> (1,32,80,80,80), bn + prelu(a2)
    run_layer(h2, 64, 32, 20, 20, 40, k3, b3, g3, be3, a2, w20,
              wigS1, 40, wigS2, 80, wigG128, y3);
    // integrate gamma: (1,32,80,80)
    { int tot = 32 * 80 * 80;
      k_sumlast<<<(tot + 255) / 256, 256, 0, stream>>>(y3, outp, tot, 80); }
}